// MatMul_18648747999367
// MI455X (gfx1250) — compile-verified
//
#include <hip/hip_runtime.h>

typedef __attribute__((ext_vector_type(2))) float v2f;
typedef __attribute__((ext_vector_type(8))) float v8f;

#define BATCH 24      // 2*12
#define MDIM 1024
#define NDIM 1024
#define KDIM 64

// One wave computes a 16(M) x 64(N) strip of C for one batch.
// WMMA fragment layouts per CDNA5 ISA 7.12.2 (32-bit data, wave32):
//   A 16x4 :  lane l -> M = l%16 ; VGPR j -> K = j + 2*(l/16)
//   B 4x16 :  lane l -> N = l%16 ; VGPR j -> K = j + 2*(l/16)
//   C 16x16:  lane l -> N = l%16 ; VGPR r -> M = r + 8*(l/16)
__global__ __launch_bounds__(256) void bmm_wmma_f32(const float* __restrict__ A,
                                                    const float* __restrict__ B,
                                                    float* __restrict__ C) {
  const int lane = threadIdx.x & 31;
  const int waveInBlock = threadIdx.x >> 5;
  const long wave = (long)blockIdx.x * 8 + waveInBlock;

  const int m_tiles = MDIM / 16;   // 64
  const int n_strips = NDIM / 64;  // 16
  const long wavesPerBatch = (long)m_tiles * n_strips;  // 1024

  const int b  = (int)(wave / wavesPerBatch);
  const int r  = (int)(wave % wavesPerBatch);
  const int mt = r / n_strips;     // 0..63
  const int ns = r % n_strips;     // 0..15
  if (b >= BATCH) return;          // wave-uniform: EXEC stays all-1 for WMMA

  const float* __restrict__ Ab = A + (size_t)b * MDIM * KDIM;
  const float* __restrict__ Bb = B + (size_t)b * KDIM * NDIM;
  float*       __restrict__ Cb = C + (size_t)b * MDIM * NDIM;

  const int half = lane >> 4;      // 0 or 1 (lane half)
  const int l16  = lane & 15;

  const int m0 = mt * 16;
  const int n0 = ns * 64;

  v8f acc0 = {}, acc1 = {}, acc2 = {}, acc3 = {};

  // A: this lane's row, with K phase offset for its half.
  const float* __restrict__ arow = Ab + (size_t)(m0 + l16) * KDIM + 2 * half;

  #pragma unroll
  for (int k0 = 0; k0 < KDIM; k0 += 4) {
    // A fragment: VGPR0 -> K=k0+2*half, VGPR1 -> K=k0+2*half+1 (contiguous b64 load)
    v2f a = *(const v2f*)(arow + k0);

    // B fragments for the 4 N-tiles of this strip.
    const float* __restrict__ bptr = Bb + (size_t)(k0 + 2 * half) * NDIM + n0 + l16;
    v2f b0, b1, b2, b3;
    b0.x = bptr[0];      b0.y = bptr[NDIM + 0];
    b1.x = bptr[16];     b1.y = bptr[NDIM + 16];
    b2.x = bptr[32];     b2.y = bptr[NDIM + 32];
    b3.x = bptr[48];     b3.y = bptr[NDIM + 48];

    acc0 = __builtin_amdgcn_wmma_f32_16x16x4_f32(false, a, false, b0, (short)0, acc0, false, false);
    acc1 = __builtin_amdgcn_wmma_f32_16x16x4_f32(false, a, false, b1, (short)0, acc1, false, false);
    acc2 = __builtin_amdgcn_wmma_f32_16x16x4_f32(false, a, false, b2, (short)0, acc2, false, false);
    acc3 = __builtin_amdgcn_wmma_f32_16x16x4_f32(false, a, false, b3, (short)0, acc3, false, false);
  }

  // Store: lane half picks M base (m0 or m0+8); each VGPR row r is M = base + r.
  float* __restrict__ crow = Cb + (size_t)(m0 + 8 * half) * NDIM + n0 + l16;
  #pragma unroll
  for (int rr = 0; rr < 8; ++rr) {
    float* __restrict__ cr = crow + (size_t)rr * NDIM;
    cr[0]  = acc0[rr];
    cr[16] = acc1[rr];
    cr[32] = acc2[rr];
    cr[48] = acc3[rr];
  }
}

extern "C" void kernel_launch(void* const* d_in, const int* in_sizes, int n_in,
                              void* d_out, int out_size, void* d_ws, size_t ws_size,
                              hipStream_t stream) {
  (void)in_sizes; (void)n_in; (void)d_ws; (void)ws_size; (void)out_size;
  const float* A = (const float*)d_in[0];   // x1 [2,12,1024,64]
  const float* B = (const float*)d_in[1];   // x2 [2,12,64,1024]
  float* C = (float*)d_out;                 // [2,12,1024,1024]

  // 24 batches * 64 m-tiles * 16 n-strips = 24576 waves; 8 waves / 256-thread block
  const int totalWaves = BATCH * (MDIM / 16) * (NDIM / 64);
  const int blocks = totalWaves / 8;        // 3072
  bmm_wmma_f32<<<blocks, 256, 0, stream>>>(A, B, C);
}